// SNN_23527830848088
// MI455X (gfx1250) — compile-verified
//
#include <hip/hip_runtime.h>

// SNN step for MI455X (gfx1250, wave32).
// current = Xd[7] @ W_int + Xext @ W_ext via V_WMMA_F32_16X16X4_F32 (exact f32).
// Memory-bound: 402 MB of weights streamed once; floor ~17us @ 23.3 TB/s.

typedef __attribute__((ext_vector_type(2))) float v2f;
typedef __attribute__((ext_vector_type(8))) float v8f;

#define BB   16
#define NN   8192
#define NIN  4096
#define BN   (BB * NN)        // 131072 elements per [B,N] plane
#define KS_INT 8
#define KS_EXT 4
#define KCHUNK 1024

// decay constants: exp(-1e-3/0.02), exp(-1e-3/0.2)
#define ALPHA_C   0.95122942450071400f
#define ONE_MALPHA 0.04877057549928600f
#define RHO_A_C   0.99501247919268232f

__global__ void snn_zero_cur(float* __restrict__ cur) {
  int i = blockIdx.x * blockDim.x + threadIdx.x;
  if (i < BN) cur[i] = 0.0f;
}

__global__ __launch_bounds__(32)
void snn_gemm_wmma(const float* __restrict__ Xd7,   // [16, 8192] delayed spikes
                   const float* __restrict__ Xext,  // [16, 4096] external spikes
                   const float* __restrict__ Wint,  // [8192, 8192]
                   const float* __restrict__ Wext,  // [4096, 8192]
                   float* __restrict__ cur)         // [16, 8192] accumulator
{
  const int lane = threadIdx.x;          // one wave32 per block
  const int n0   = blockIdx.x * 32;      // 32-column output tile (2 WMMA tiles)
  const int ks   = blockIdx.y;           // K split: 0..7 internal, 8..11 external

  const float* Asrc;
  const float* W;
  int k0, ldA;
  if (ks < KS_INT) { Asrc = Xd7;  W = Wint; k0 = ks * KCHUNK;            ldA = NN;  }
  else             { Asrc = Xext; W = Wext; k0 = (ks - KS_INT) * KCHUNK; ldA = NIN; }

  // ISA f32 WMMA operand layouts (16x16x4):
  //  A 16x4 : lanes 0-15 -> M=lane, K=0/1 in v0/v1 ; lanes 16-31 -> K=2/3
  //  B 4x16 : lanes 0-15 -> N=lane, K=0/1 in v0/v1 ; lanes 16-31 -> K=2/3
  //  C 16x16: VGPR r -> M = r + 8*(lane>>4), N = n0 + (lane&15)
  const int half = lane >> 4;            // 0 or 1
  const int idx  = lane & 15;            // M row for A, N col for B
  const int khi  = half * 2;

  const float* aPtr = Asrc + (size_t)idx * ldA + k0 + khi;       // float2-aligned
  const float* wPtr = W + (size_t)(k0 + khi) * NN + n0 + idx;

  v8f c0 = {};
  v8f c1 = {};

  for (int kk = 0; kk < KCHUNK; kk += 4) {
    v2f a = *(const v2f*)aPtr;           // A[idx, k+khi .. k+khi+1]
    v2f b0, b1;
    b0.x = wPtr[0];                      // W[k+khi+0, n0+idx]
    b0.y = wPtr[NN];                     // W[k+khi+1, n0+idx]
    b1.x = wPtr[16];                     // second 16-col tile
    b1.y = wPtr[NN + 16];
    __builtin_prefetch(wPtr + (size_t)16 * NN, 0, 3);  // stream weights ahead
    c0 = __builtin_amdgcn_wmma_f32_16x16x4_f32(false, a, false, b0,
                                               (short)0, c0, false, false);
    c1 = __builtin_amdgcn_wmma_f32_16x16x4_f32(false, a, false, b1,
                                               (short)0, c1, false, false);
    aPtr += 4;
    wPtr += (size_t)4 * NN;
  }

  // Accumulate K-split partials into cur[16,8192] with f32 atomics.
  float* base = cur + n0 + idx;
  const int m8 = half * 8;
#pragma unroll
  for (int r = 0; r < 8; ++r) {
    atomicAdd(base + (size_t)(m8 + r) * NN,      c0[r]);
    atomicAdd(base + (size_t)(m8 + r) * NN + 16, c1[r]);
  }
}

__global__ void snn_pointwise(const float* __restrict__ V,
                              const float* __restrict__ a,
                              const float* __restrict__ Xd,   // [8,16,8192]
                              const float* __restrict__ cur,  // [16,8192]
                              float* __restrict__ out)        // X|V_new|a_new|Xd_new
{
  int i = blockIdx.x * blockDim.x + threadIdx.x;
  if (i >= BN) return;
  float av = a[i];
  float vv = V[i];
  float th = 1.0f + 1.8f * av;
  float x  = (vv >= th) ? 1.0f : 0.0f;
  float anew = RHO_A_C * av + x;
  float vnew = ALPHA_C * vv * (1.0f - x) + ONE_MALPHA * cur[i];

  out[i]          = x;       // X
  out[BN + i]     = vnew;    // V_new
  out[2 * BN + i] = anew;    // a_new

  float* xdn = out + 3 * BN; // Xd_new [8,16,8192]
  xdn[i] = x;                // slot 0 <- new spikes
#pragma unroll
  for (int d = 1; d < 8; ++d)
    xdn[d * BN + i] = Xd[(d - 1) * BN + i];   // shift delay buffer
}

extern "C" void kernel_launch(void* const* d_in, const int* in_sizes, int n_in,
                              void* d_out, int out_size, void* d_ws, size_t ws_size,
                              hipStream_t stream) {
  const float* V    = (const float*)d_in[0];
  const float* a    = (const float*)d_in[1];
  const float* Xd   = (const float*)d_in[2];   // [8,16,8192]
  const float* Xext = (const float*)d_in[3];   // [16,4096]
  const float* Wint = (const float*)d_in[4];   // [8192,8192]
  const float* Wext = (const float*)d_in[5];   // [4096,8192]
  float* out = (float*)d_out;
  float* cur = (float*)d_ws;                   // 512 KB scratch: current[16,8192]

  const float* Xd7 = Xd + 7 * BN;              // Xd[-1]

  snn_zero_cur<<<dim3((BN + 255) / 256), dim3(256), 0, stream>>>(cur);
  snn_gemm_wmma<<<dim3(NN / 32, KS_INT + KS_EXT), dim3(32), 0, stream>>>(
      Xd7, Xext, Wint, Wext, cur);
  snn_pointwise<<<dim3((BN + 255) / 256), dim3(256), 0, stream>>>(
      V, a, Xd, cur, out);
}